// DeformConv3D_alternative_27822798143505
// MI455X (gfx1250) — compile-verified
//
#include <hip/hip_runtime.h>

// Deformable 3D conv (B=1, C=16, S=32, ks=3, pad=1), fused single kernel.
//   Sampling: corners OUTER, channels INNER. Each corner-pair issues 16 loads
//   off ONE address VGPR pair (channel stride 128KB encoded in the 24-bit
//   immediate offset) into 16 distinct temps -> one s_clause + one wait,
//   then 16 v_fmac into 8 independent accumulators. x is 2MB => L2-resident.
//   Contraction: out[o,pos] = sum_{c,n} W[o,c,n]*S[c,n,pos] via
//   V_WMMA_F32_16X16X4_F32 (exact fp32), K = 16*27 = 432 in 108 steps/wave.
//   Weights (27KB) staged transposed in LDS as lds_w[n][c][o].

typedef float v2f __attribute__((ext_vector_type(2)));
typedef float v8f __attribute__((ext_vector_type(8)));

#define SP   32768          // 32^3 positions
#define NC   16             // channels (in = out = 16)
#define NK   27             // kernel taps (3^3)

// two corners, 16 independent loads (same base VGPR pair, immediate channel
// offsets), then fmacs into 8 independent accumulators
#define GATHER2C(oA, wA, oB, wB) do {                                          \
    const float* pA_ = xh + (oA);                                              \
    const float* pB_ = xh + (oB);                                              \
    const float a0_ = pA_[ 0 << 15], a1_ = pA_[ 1 << 15];                      \
    const float a2_ = pA_[ 4 << 15], a3_ = pA_[ 5 << 15];                      \
    const float a4_ = pA_[ 8 << 15], a5_ = pA_[ 9 << 15];                      \
    const float a6_ = pA_[12 << 15], a7_ = pA_[13 << 15];                      \
    const float b0_ = pB_[ 0 << 15], b1_ = pB_[ 1 << 15];                      \
    const float b2_ = pB_[ 4 << 15], b3_ = pB_[ 5 << 15];                      \
    const float b4_ = pB_[ 8 << 15], b5_ = pB_[ 9 << 15];                      \
    const float b6_ = pB_[12 << 15], b7_ = pB_[13 << 15];                      \
    s0 = fmaf(wA, a0_, fmaf(wB, b0_, s0));                                     \
    s1 = fmaf(wA, a1_, fmaf(wB, b1_, s1));                                     \
    s2 = fmaf(wA, a2_, fmaf(wB, b2_, s2));                                     \
    s3 = fmaf(wA, a3_, fmaf(wB, b3_, s3));                                     \
    s4 = fmaf(wA, a4_, fmaf(wB, b4_, s4));                                     \
    s5 = fmaf(wA, a5_, fmaf(wB, b5_, s5));                                     \
    s6 = fmaf(wA, a6_, fmaf(wB, b6_, s6));                                     \
    s7 = fmaf(wA, a7_, fmaf(wB, b7_, s7));                                     \
} while (0)

__global__ __launch_bounds__(256)
void deform_conv3d_wmma_f32(const float* __restrict__ x,     // [16][32][32][32]
                            const float* __restrict__ off,   // [81][32][32][32]
                            const float* __restrict__ wgt,   // [16][16][3][3][3]
                            float* __restrict__ out)         // [16][32][32][32]
{
    __shared__ float lds_w[NK * NC * NC];      // [n][c][o], 27*256 floats = 27KB

    const int tid = threadIdx.x;
    // cooperative transpose of weights into LDS: lds_w[n*256 + c*16 + o] = W[o][c][n]
    for (int i = tid; i < NK * 256; i += 256) {
        const int n = i >> 8, rem = i & 255, c = rem >> 4, o = rem & 15;
        lds_w[i] = wgt[(o * NC + c) * NK + n];
    }
    __syncthreads();

    const int wave = tid >> 5;
    const int lane = tid & 31;
    const int j    = lane & 15;          // position-in-tile / output-channel row index
    const int hi   = lane >> 4;          // upper half-wave flag
    const int hi2  = hi << 1;            // channel offset pattern for A/B fragments

    const int pos = blockIdx.x * 128 + wave * 16 + j;   // this lane's output voxel
    const int h = pos >> 10, w = (pos >> 5) & 31, d = pos & 31;

    const float* xh = x + (hi2 << 15);   // fold lane's channel-group base into address

    v8f acc = {0.f, 0.f, 0.f, 0.f, 0.f, 0.f, 0.f, 0.f};

    for (int n = 0; n < NK; ++n) {
        // kernel-relative offsets in {-1,0,1}, n = i*9 + j*3 + k ordering
        const int rx = n / 9 - 1;
        const int ry = (n / 3) % 3 - 1;
        const int rz = n % 3 - 1;

        // learned offsets: channels [0..26]=x, [27..53]=y, [54..80]=z
        const float ox = off[(0 * NK + n) * SP + pos];
        const float oy = off[(1 * NK + n) * SP + pos];
        const float oz = off[(2 * NK + n) * SP + pos];

        // sampling position in padded coords (base = idx+1), padded size 34
        const float px = (float)(h + 1 + rx) + ox;
        const float py = (float)(w + 1 + ry) + oy;
        const float pz = (float)(d + 1 + rz) + oz;

        // --- per-axis floor/clamp + reference snap rule + trilinear weights ---
        const float fx = floorf(px), fy = floorf(py), fz = floorf(pz);
        int q0x = min(max((int)fx, 0), 33), q1x = min(max((int)fx + 1, 0), 33);
        int q0y = min(max((int)fy, 0), 33), q1y = min(max((int)fy + 1, 0), 33);
        int q0z = min(max((int)fz, 0), 33), q1z = min(max((int)fz + 1, 0), 33);

        float sx = (px < 1.f || px > 32.f) ? fx : px; sx = fminf(fmaxf(sx, 0.f), 33.f);
        float sy = (py < 1.f || py > 32.f) ? fy : py; sy = fminf(fmaxf(sy, 0.f), 33.f);
        float sz = (pz < 1.f || pz > 32.f) ? fz : pz; sz = fminf(fmaxf(sz, 0.f), 33.f);

        float lx = 1.f + ((float)q0x - sx), hx = 1.f - ((float)q1x - sx);
        float ly = 1.f + ((float)q0y - sy), hy = 1.f - ((float)q1y - sy);
        float lz = 1.f + ((float)q0z - sz), hz = 1.f - ((float)q1z - sz);

        // fold zero-padding validity into the axis weights (branch-free, EXEC stays full)
        lx *= (q0x >= 1 && q0x <= 32) ? 1.f : 0.f;
        hx *= (q1x >= 1 && q1x <= 32) ? 1.f : 0.f;
        ly *= (q0y >= 1 && q0y <= 32) ? 1.f : 0.f;
        hy *= (q1y >= 1 && q1y <= 32) ? 1.f : 0.f;
        lz *= (q0z >= 1 && q0z <= 32) ? 1.f : 0.f;
        hz *= (q1z >= 1 && q1z <= 32) ? 1.f : 0.f;

        // clamped unpadded gather indices (loads always in-bounds)
        const int i0x = min(max(q0x - 1, 0), 31) << 10, i1x = min(max(q1x - 1, 0), 31) << 10;
        const int i0y = min(max(q0y - 1, 0), 31) << 5,  i1y = min(max(q1y - 1, 0), 31) << 5;
        const int i0z = min(max(q0z - 1, 0), 31),       i1z = min(max(q1z - 1, 0), 31);

        const int o000 = i0x + i0y + i0z, o001 = i0x + i0y + i1z;
        const int o010 = i0x + i1y + i0z, o011 = i0x + i1y + i1z;
        const int o100 = i1x + i0y + i0z, o101 = i1x + i0y + i1z;
        const int o110 = i1x + i1y + i0z, o111 = i1x + i1y + i1z;

        const float w000 = lx * ly * lz, w001 = lx * ly * hz;
        const float w010 = lx * hy * lz, w011 = lx * hy * hz;
        const float w100 = hx * ly * lz, w101 = hx * ly * hz;
        const float w110 = hx * hy * lz, w111 = hx * hy * hz;

        // --- trilinear sample, corners outer / channels inner:
        //     s0..s7 hold this lane's channels {0,1,4,5,8,9,12,13}+hi2 ---
        float s0 = 0.f, s1 = 0.f, s2 = 0.f, s3 = 0.f;
        float s4 = 0.f, s5 = 0.f, s6 = 0.f, s7 = 0.f;
        GATHER2C(o000, w000, o001, w001);
        GATHER2C(o010, w010, o011, w011);
        GATHER2C(o100, w100, o101, w101);
        GATHER2C(o110, w110, o111, w111);

        // --- 4 WMMA steps per tap: K rows are channels 4t..4t+3 ---
        const float* wl = lds_w + n * 256;
        {
            v2f A, B;
            A.x = wl[(0 + hi2) * NC + j];  A.y = wl[(1 + hi2) * NC + j];
            B.x = s0;                      B.y = s1;
            acc = __builtin_amdgcn_wmma_f32_16x16x4_f32(false, A, false, B,
                                                        (short)0, acc, false, false);
            A.x = wl[(4 + hi2) * NC + j];  A.y = wl[(5 + hi2) * NC + j];
            B.x = s2;                      B.y = s3;
            acc = __builtin_amdgcn_wmma_f32_16x16x4_f32(false, A, false, B,
                                                        (short)0, acc, false, false);
            A.x = wl[(8 + hi2) * NC + j];  A.y = wl[(9 + hi2) * NC + j];
            B.x = s4;                      B.y = s5;
            acc = __builtin_amdgcn_wmma_f32_16x16x4_f32(false, A, false, B,
                                                        (short)0, acc, false, false);
            A.x = wl[(12 + hi2) * NC + j]; A.y = wl[(13 + hi2) * NC + j];
            B.x = s6;                      B.y = s7;
            acc = __builtin_amdgcn_wmma_f32_16x16x4_f32(false, A, false, B,
                                                        (short)0, acc, false, false);
        }
    }

    // D tile store: VGPR r holds rows M=r (lanes 0-15) / M=r+8 (lanes 16-31), col = j
#pragma unroll
    for (int r = 0; r < 8; ++r)
        out[(r + (hi << 3)) * SP + pos] = acc[r];
}

extern "C" void kernel_launch(void* const* d_in, const int* in_sizes, int n_in,
                              void* d_out, int out_size, void* d_ws, size_t ws_size,
                              hipStream_t stream) {
    const float* x   = (const float*)d_in[0];   // [1,16,32,32,32]
    const float* off = (const float*)d_in[1];   // [1,81,32,32,32]
    const float* wgt = (const float*)d_in[2];   // [16,16,3,3,3]
    float*       out = (float*)d_out;           // [1,16,32,32,32]

    // 8 waves/block * 16 positions/wave = 128 positions per block; 32768/128 = 256 blocks
    deform_conv3d_wmma_f32<<<dim3(SP / 128), 256, 0, stream>>>(x, off, wgt, out);
}